// mLSTMCell_35837207117872
// MI455X (gfx1250) — compile-verified
//
#include <hip/hip_runtime.h>
#include <math.h>

typedef __attribute__((ext_vector_type(2))) float v2f;
typedef __attribute__((ext_vector_type(4))) float v4f;
typedef __attribute__((ext_vector_type(8))) float v8f;

#define H   1024
#define IN_ 1024
#define B   64
#define BH  (B * H)  // 65536

// ---------------------------------------------------------------------------
// Kernel 1: six fused projections via V_WMMA_F32_16X16X4_F32.
// out[b,h] = sum_k x[b,k] * W[h,k]  (+bias, +nonlinearity per gate)
// A = x tile (M=batch,K=in), B = W^T tile (K=in,N=h)  => B[k][n] = W[h0+n][k]
// Fragment layout (ISA 7.12.2, 32-bit A 16x4): lanes 0-15: {K+0,K+1} of row l,
// lanes 16-31: {K+2,K+3} of row l  -> contiguous float2 per lane for A and B.
// ---------------------------------------------------------------------------
struct ProjArgs {
  const float* w[6];
  const float* bias[6];
  const float* x;
  float*       act;   // [6][B*H] : q,k,v,i,f,o
};

__global__ __launch_bounds__(32)
void mlstm_proj_wmma(ProjArgs args) {
  const int htile = blockIdx.x;   // 0..63   (H/16)
  const int btile = blockIdx.y;   // 0..3    (B/16)
  const int gate  = blockIdx.z;   // 0..5    q,k,v,i,f,o
  const int lane  = threadIdx.x;
  const int half  = lane >> 4;    // 0 or 1
  const int l16   = lane & 15;

  const float* __restrict__ xrow =
      args.x + (size_t)(btile * 16 + l16) * IN_ + half * 2;
  const float* __restrict__ wrow =
      args.w[gate] + (size_t)(htile * 16 + l16) * IN_ + half * 2;

  v8f acc = {};
#pragma unroll 8
  for (int s = 0; s < IN_ / 4; ++s) {
    v2f a = *(const v2f*)(xrow + s * 4);
    v2f b = *(const v2f*)(wrow + s * 4);
    acc = __builtin_amdgcn_wmma_f32_16x16x4_f32(
        /*neg_a=*/false, a, /*neg_b=*/false, b,
        /*c_mod=*/(short)0, acc, /*reuse_a=*/false, /*reuse_b=*/false);
  }

  // D layout: VGPR r holds M=r (lanes 0-15) and M=r+8 (lanes 16-31), N=lane&15
  const int   hcol = htile * 16 + l16;
  const float bia  = args.bias[gate][hcol];
  float* __restrict__ out = args.act + (size_t)gate * BH;
  const float inv_sqrt_d = 0.03125f;  // 1/sqrt(1024)

#pragma unroll
  for (int r = 0; r < 8; ++r) {
    const int brow = btile * 16 + half * 8 + r;
    float y = acc[r];
    switch (gate) {
      case 0: y = y + bia; break;                           // q
      case 1: y = y * inv_sqrt_d + bia; break;              // k (scaled)
      case 2: y = y + bia; break;                           // v
      case 3: y = expf(y + bia); break;                     // i (exp gate)
      default: y = 1.0f / (1.0f + expf(-(y + bia))); break; // f,o (sigmoid)
    }
    out[(size_t)brow * H + hcol] = y;
  }
}

// ---------------------------------------------------------------------------
// Kernel 2: bandwidth-bound streaming update of the matrix memory, fused with
// the retrieval numerator so c_t is written once and never re-read:
//   c_t[b,i,j] = f[b,i]*c_prev[b,i,j] + i[b,i]*v[b,i]*k[b,j]
//   num[b,i]   = sum_j c_t[b,i,j] * q[b,j]
// One 256-thread block per (b,i) row; each thread owns one v4f (1024/4=256).
// c_prev / c_t are 268 MB each, touched exactly once -> non-temporal hints so
// the stream does not thrash L2 and evict the hot q/k/activation rows.
// ---------------------------------------------------------------------------
__global__ __launch_bounds__(256)
void mlstm_cell_update(const float* __restrict__ c_prev,
                       const float* __restrict__ act,
                       float* __restrict__ c_out,
                       float* __restrict__ num) {
  const int row = blockIdx.x;      // b*H + i
  const int b   = row >> 10;

  const float fg = act[4 * BH + row];
  const float iv = act[3 * BH + row] * act[2 * BH + row];  // i * v

  const v4f* __restrict__ cp  = (const v4f*)(c_prev + (size_t)row * H);
  v4f*       __restrict__ co  = (v4f*)(c_out + (size_t)row * H);
  const v4f* __restrict__ k4p = (const v4f*)(act + 1 * BH + (size_t)b * H);
  const v4f* __restrict__ q4p = (const v4f*)(act + 0 * BH + (size_t)b * H);

  const int t = threadIdx.x;
  const v4f c0 = __builtin_nontemporal_load(cp + t);  // stream: read once
  const v4f k0 = k4p[t];                              // hot: keep cached
  const v4f q0 = q4p[t];                              // hot: keep cached

  const v4f r = fg * c0 + iv * k0;
  __builtin_nontemporal_store(r, co + t);             // stream: write once

  const v4f pq = r * q0;
  float partial = pq.x + pq.y + pq.z + pq.w;

  __shared__ float red[256];
  red[t] = partial;
  __syncthreads();
#pragma unroll
  for (int s = 128; s > 0; s >>= 1) {
    if (t < s) red[t] += red[t + s];
    __syncthreads();
  }
  if (t == 0) num[row] = red[0];
}

// ---------------------------------------------------------------------------
// Kernel 3: n_t, denominator, and h_t. One block per batch element.
//   n_t = f*n_prev + i*k ;  den = max(|n_t . q|, 1) ;  h = o * num / den
// ---------------------------------------------------------------------------
__global__ __launch_bounds__(256)
void mlstm_finalize(const float* __restrict__ n_prev,
                    const float* __restrict__ act,
                    const float* __restrict__ num,
                    float* __restrict__ h_out,
                    float* __restrict__ n_out) {
  const int b = blockIdx.x;
  const int t = threadIdx.x;

  const float* __restrict__ qv = act + 0 * BH + (size_t)b * H;
  const float* __restrict__ kv = act + 1 * BH + (size_t)b * H;
  const float* __restrict__ ig = act + 3 * BH + (size_t)b * H;
  const float* __restrict__ fg = act + 4 * BH + (size_t)b * H;
  const float* __restrict__ og = act + 5 * BH + (size_t)b * H;
  const float* __restrict__ np = n_prev + (size_t)b * H;

  float partial = 0.0f;
#pragma unroll
  for (int u = 0; u < 4; ++u) {
    const int idx = t + u * 256;
    const float n = fg[idx] * np[idx] + ig[idx] * kv[idx];
    n_out[(size_t)b * H + idx] = n;
    partial += n * qv[idx];
  }

  __shared__ float red[256];
  __shared__ float den_s;
  red[t] = partial;
  __syncthreads();
#pragma unroll
  for (int s = 128; s > 0; s >>= 1) {
    if (t < s) red[t] += red[t + s];
    __syncthreads();
  }
  if (t == 0) den_s = fmaxf(fabsf(red[0]), 1.0f);
  __syncthreads();

  const float inv_den = 1.0f / den_s;
#pragma unroll
  for (int u = 0; u < 4; ++u) {
    const int idx = t + u * 256;
    h_out[(size_t)b * H + idx] = og[idx] * num[(size_t)b * H + idx] * inv_den;
  }
}

// ---------------------------------------------------------------------------
// Launch. Inputs (setup_inputs order):
//  0:x_t 1:c_prev 2:n_prev 3..8:w_q,w_k,w_v,w_i,w_f,w_o 9..14:b_q..b_o
// Output (concat): h_t [65536], c_t [67108864], n_t [65536]
// Workspace: act 6*BH floats + num BH floats  (~1.8 MB)
// ---------------------------------------------------------------------------
extern "C" void kernel_launch(void* const* d_in, const int* in_sizes, int n_in,
                              void* d_out, int out_size, void* d_ws, size_t ws_size,
                              hipStream_t stream) {
  (void)in_sizes; (void)n_in; (void)out_size; (void)ws_size;

  const float* x_t    = (const float*)d_in[0];
  const float* c_prev = (const float*)d_in[1];
  const float* n_prev = (const float*)d_in[2];

  float* ws  = (float*)d_ws;
  float* act = ws;             // [6][BH]
  float* num = ws + 6 * BH;    // [BH]

  float* h_out = (float*)d_out;
  float* c_out = h_out + BH;
  float* n_out = c_out + (size_t)B * H * H;

  ProjArgs pa;
  for (int g = 0; g < 6; ++g) {
    pa.w[g]    = (const float*)d_in[3 + g];
    pa.bias[g] = (const float*)d_in[9 + g];
  }
  pa.x   = x_t;
  pa.act = act;

  // 1) six projections: grid = (H/16, B/16, 6), one wave per 16x16 tile
  mlstm_proj_wmma<<<dim3(H / 16, B / 16, 6), 32, 0, stream>>>(pa);

  // 2) streaming matrix-memory update + fused retrieval numerator (NT stream)
  mlstm_cell_update<<<BH, 256, 0, stream>>>(c_prev, act, c_out, num);

  // 3) normalizer state, denominator, output gate
  mlstm_finalize<<<B, 256, 0, stream>>>(n_prev, act, num, h_out, n_out);
}